// Model_65901978190419
// MI455X (gfx1250) — compile-verified
//
#include <hip/hip_runtime.h>
#include <math.h>

// ---------------- static grid-encoder configuration ----------------
#define N_RAYS     32768
#define N_MULTI    6
#define NUM_LEVELS 10
#define LEVEL_DIM  4
#define FEAT_DIM   40          // NUM_LEVELS * LEVEL_DIM

__constant__ int c_res[NUM_LEVELS] = {16,32,64,128,256,512,1024,2048,4096,8192};
__constant__ int c_off[NUM_LEVELS] = {0,4920,40864,315496,2412648,4509800,
                                      6606952,8704104,10801256,12898408};
// levels 0..2 dense ((R+1)^3 entries), levels 3..9 hashed, table size 2^21
#define HASH_MASK 0x1FFFFFu
#define PRIME_Y 2654435761u
#define PRIME_Z 805459861u

// ======================================================================
// Kernel 1: contraction + multilevel hash-grid encode + erf weighting.
// ======================================================================
__global__ __launch_bounds__(256)
void grid_encode_kernel(const float* __restrict__ means,   // [N,6,3]
                        const float* __restrict__ stds,    // [N,6]
                        const float* __restrict__ emb,     // [TOTAL,4]
                        float* __restrict__ feat_out)      // [N,40]
{
    int t = blockIdx.x * blockDim.x + threadIdx.x;
    if (t >= N_RAYS * NUM_LEVELS) return;
    const int r = t / NUM_LEVELS;
    const int l = t % NUM_LEVELS;

    const int   R    = c_res[l];
    const bool  hash = (l >= 3);
    const float Rm1  = (float)(R - 1);
    const float Rf   = (float)R;
    const unsigned Rp = (unsigned)(R + 1);
    const float4* __restrict__ tab = (const float4*)emb + c_off[l];

    float acc0 = 0.f, acc1 = 0.f, acc2 = 0.f, acc3 = 0.f;

    for (int s = 0; s < N_MULTI; ++s) {
        const float* m = means + ((size_t)r * N_MULTI + s) * 3;
        float x = m[0], y = m[1], z = m[2];
        float sd = stds[(size_t)r * N_MULTI + s];

        float r2 = fmaxf(x*x + y*y + z*z, 1.1920929e-07f);
        float rr = sqrtf(r2);
        if (r2 > 1.0f) {
            float sc = (2.f * rr - 1.f) / r2;
            x *= sc; y *= sc; z *= sc;
            float det = (1.f / r2) * sc * sc;
            sd *= cbrtf(det);
        }
        x *= 0.5f; y *= 0.5f; z *= 0.5f; sd *= 0.5f;
        float ux = fminf(fmaxf((x + 1.f) * 0.5f, 0.f), 1.f);
        float uy = fminf(fmaxf((y + 1.f) * 0.5f, 0.f), 1.f);
        float uz = fminf(fmaxf((z + 1.f) * 0.5f, 0.f), 1.f);

        float px = ux * Rm1 + 0.5f, py = uy * Rm1 + 0.5f, pz = uz * Rm1 + 0.5f;
        float flx = floorf(px), fly = floorf(py), flz = floorf(pz);
        float fx = px - flx, fy = py - fly, fz = pz - flz;
        unsigned ix = (unsigned)flx, iy = (unsigned)fly, iz = (unsigned)flz;

        float f0 = 0.f, f1 = 0.f, f2 = 0.f, f3 = 0.f;
        #pragma unroll
        for (int c = 0; c < 8; ++c) {
            unsigned cx = ix + ((c >> 2) & 1);
            unsigned cy = iy + ((c >> 1) & 1);
            unsigned cz = iz + (c & 1);
            unsigned idx;
            if (hash) idx = (cx ^ (cy * PRIME_Y) ^ (cz * PRIME_Z)) & HASH_MASK;
            else      idx = cx + cy * Rp + cz * Rp * Rp;
            float wc = ((c & 4) ? fx : 1.f - fx) *
                       ((c & 2) ? fy : 1.f - fy) *
                       ((c & 1) ? fz : 1.f - fz);
            float4 e = tab[idx];
            f0 = fmaf(wc, e.x, f0); f1 = fmaf(wc, e.y, f1);
            f2 = fmaf(wc, e.z, f2); f3 = fmaf(wc, e.w, f3);
        }
        float w = erff(1.0f / fmaxf(sqrtf(8.f * sd * sd * Rf * Rf), 1e-10f));
        acc0 = fmaf(w, f0, acc0); acc1 = fmaf(w, f1, acc1);
        acc2 = fmaf(w, f2, acc2); acc3 = fmaf(w, f3, acc3);
    }
    const float inv = 1.f / (float)N_MULTI;
    float* o = feat_out + (size_t)r * FEAT_DIM + l * LEVEL_DIM;
    o[0] = acc0 * inv; o[1] = acc1 * inv; o[2] = acc2 * inv; o[3] = acc3 * inv;
}

// ======================================================================
// Kernel 2: fused MLP, V_WMMA_F32_16X16X4_F32 + async-LDS double-buffered
// weight staging. One workgroup (8 wave32) per 32-ray tile.
// ======================================================================
typedef __attribute__((ext_vector_type(2))) float v2f;
typedef __attribute__((ext_vector_type(8))) float v8f;

#define B2LD 544     // buf2 row stride: [0..255]=y, [256..511]=x, [512..538]=dir, pad->0

__device__ __forceinline__ v8f wmma4(v2f a, v2f b, v8f c) {
    return __builtin_amdgcn_wmma_f32_16x16x4_f32(false, a, false, b,
                                                 (short)0, c, false, false);
}

// D[.,NOUT-slice] = act(A[.,K] @ W[K,NOUT] + bias).  TJ = 16-wide N strips
// per wave, TM = 16-row M tiles per wave (B fragment reused across TM).
// Static tile shape => accumulators stay in fixed registers, uniform flow.
template<int NOUT, int TJ, int TM>
__device__ __forceinline__ void gemm_layer(
    const float* sA, int lda, int aOff, int K,
    const float* __restrict__ W, const float* __restrict__ bias,
    float* sW0, float* sW1, float* sDump,
    float* sD, int ldd, int dOff, bool relu,
    int tid, int n0base, int mRowBase)
{
    const int lane = tid & 31;
    const int l16  = lane & 15;
    const int half = lane >> 4;
    const int K4   = (K + 3) & ~3;
    const int P    = (K4 + 15) >> 4;          // 16-row K panels
    constexpr int N4P = (16 * NOUT) / 4;      // float4s per panel

    v8f acc[TJ][TM] = {};

    // Stage panel p (rows p*16.. of W) into buf; tail rows -> dump + zeros.
    auto stage = [&](int p, float* buf) {
        const int kp = p * 16;
        int valid = K - kp; if (valid > 16) valid = 16;
        for (int i = tid; i < N4P; i += 256) {
            int row = i / (NOUT / 4);
            unsigned loff;
            const float* g;
            if (row < valid) {
                loff = (unsigned)(uintptr_t)(buf + 4 * i);
                g    = W + (size_t)kp * NOUT + 4 * i;
            } else {                       // uniform issue count, no OOB reads
                loff = (unsigned)(uintptr_t)sDump;
                g    = W;
            }
            asm volatile("global_load_async_to_lds_b128 %0, %1, off"
                         :: "v"(loff), "v"(g) : "memory");
        }
        for (int i = valid * NOUT + tid; i < 16 * NOUT; i += 256)
            buf[i] = 0.f;
    };

    auto body = [&](const float* cur, int kp, int k0) {
        const int ka = k0 + 2 * half;
        v2f a[TM];
        #pragma unroll
        for (int m = 0; m < TM; ++m) {
            const int r = mRowBase + m * 16 + l16;
            a[m].x = sA[r * lda + aOff + kp + ka];
            a[m].y = sA[r * lda + aOff + kp + ka + 1];
        }
        #pragma unroll
        for (int j = 0; j < TJ; ++j) {
            const int n0 = n0base + j * 16;
            v2f b;
            b.x = cur[ka * NOUT + n0 + l16];
            b.y = cur[(ka + 1) * NOUT + n0 + l16];
            #pragma unroll
            for (int m = 0; m < TM; ++m)
                acc[j][m] = wmma4(a[m], b, acc[j][m]);
        }
    };

    stage(0, sW0);
    for (int p = 0; p < P; ++p) {
        float* cur = (p & 1) ? sW1 : sW0;
        float* nxt = (p & 1) ? sW0 : sW1;
        if (p + 1 < P) {
            stage(p + 1, nxt);
            // wait for panel p's copies; next panel's stay in flight
            if constexpr (NOUT == 256) asm volatile("s_wait_asynccnt 0x4" ::: "memory");
            else                       asm volatile("s_wait_asynccnt 0x1" ::: "memory");
        } else {
            asm volatile("s_wait_asynccnt 0x0" ::: "memory");
        }
        __syncthreads();                   // panel visible to all waves

        const int kp = p * 16;
        int kend = K4 - kp; if (kend > 16) kend = 16;
        if (kend == 16) {
            #pragma unroll
            for (int k0 = 0; k0 < 16; k0 += 4) body(cur, kp, k0);
        } else {
            for (int k0 = 0; k0 < kend; k0 += 4) body(cur, kp, k0);
        }
        __syncthreads();                   // done with cur before re-staging it
    }

    #pragma unroll
    for (int j = 0; j < TJ; ++j) {
        const int n0 = n0base + j * 16;
        const float bv = bias[n0 + l16];
        #pragma unroll
        for (int m = 0; m < TM; ++m) {
            #pragma unroll
            for (int v = 0; v < 8; ++v) {
                const int row = mRowBase + m * 16 + v + 8 * half;
                float x = acc[j][m][v] + bv;
                if (relu) x = fmaxf(x, 0.f);
                sD[row * ldd + dOff + n0 + l16] = x;
            }
        }
    }
}

#define SMEM_FLOATS (32*B2LD + 32*256 + 32*FEAT_DIM + 16*256 + 16*256 + 4)

__global__ __launch_bounds__(256)
void nerf_mlp_kernel(const float* __restrict__ feat,      // [N,40]
                     const float* __restrict__ viewdirs,  // [N,3]
                     const float* __restrict__ dw0, const float* __restrict__ db0,
                     const float* __restrict__ dw1, const float* __restrict__ db1,
                     const float* __restrict__ sw0, const float* __restrict__ sb0,
                     const float* __restrict__ sw1, const float* __restrict__ sb1,
                     const float* __restrict__ rw,  const float* __restrict__ rb,
                     float* __restrict__ out)              // [N] dens ++ [N,3] rgb
{
    extern __shared__ float smem[];
    float* sBuf2 = smem;                   // 32 x 544
    float* sBufY = sBuf2 + 32 * B2LD;      // 32 x 256 (h0 early, final y later)
    float* sBufF = sBufY + 32 * 256;       // 32 x 40
    float* sW0   = sBufF + 32 * FEAT_DIM;  // 16 x 256 weight panel A
    float* sW1   = sW0   + 16 * 256;       // 16 x 256 weight panel B
    float* sDump = sW1   + 16 * 256;       // async dump slot (16B)

    const int tid  = threadIdx.x;
    const int wave = tid >> 5;
    const int rayBase = blockIdx.x * 32;

    // ---- stage features, viewdir posenc, zero pad cols ----
    for (int i = tid; i < 32 * FEAT_DIM; i += 256)
        sBufF[i] = feat[(size_t)(rayBase + i / FEAT_DIM) * FEAT_DIM + (i % FEAT_DIM)];
    for (int i = tid; i < 32 * 27; i += 256) {
        int row = i / 27, d = i % 27;
        const float* v = viewdirs + (size_t)(rayBase + row) * 3;
        float val;
        if (d < 3)       val = v[d];
        else if (d < 15) { int q = d - 3;  val = sinf(v[q % 3] * (float)(1 << (q / 3))); }
        else             { int q = d - 15; val = cosf(v[q % 3] * (float)(1 << (q / 3))); }
        sBuf2[row * B2LD + 512 + d] = val;
    }
    for (int i = tid; i < 32 * 5; i += 256)
        sBuf2[(i / 5) * B2LD + 539 + (i % 5)] = 0.f;
    __syncthreads();

    // ---- L0: h0 = relu(feat @ dw0 + db0) [32,64] -> sBufY ----
    // waves 0-3: rows 0-15, cols wave*16; waves 4-7: rows 16-31, cols (wave-4)*16
    gemm_layer<64, 1, 1>(sBufF, FEAT_DIM, 0, 40, dw0, db0, sW0, sW1, sDump,
                         sBufY, 256, 0, true, tid,
                         (wave & 3) * 16, (wave >> 2) * 16);
    __syncthreads();

    // ---- L1: x = h0 @ dw1 + db1 [32,256] -> buf2 cols 256.. ----
    gemm_layer<256, 2, 2>(sBufY, 256, 0, 64, dw1, db1, sW0, sW1, sDump,
                          sBuf2, B2LD, 256, false, tid, wave * 32, 0);
    __syncthreads();

    // ---- density = softplus(x[:,0] - 1) ----
    if (tid < 32) {
        float zz = sBuf2[tid * B2LD + 256] - 1.0f;
        out[rayBase + tid] = (zz > 20.f) ? zz : log1pf(expf(zz));
    }

    // ---- L2: y = relu(inp @ sw0 + sb0), inp = cols 256..538 (K=283) ----
    gemm_layer<256, 2, 2>(sBuf2, B2LD, 256, 283, sw0, sb0, sW0, sW1, sDump,
                          sBuf2, B2LD, 0, true, tid, wave * 32, 0);
    __syncthreads();

    // ---- L3: y = relu(y2 @ sw1 + sb1), y2 = cols 0..538 (K=539) ----
    gemm_layer<256, 2, 2>(sBuf2, B2LD, 0, 539, sw1, sb1, sW0, sW1, sDump,
                          sBufY, 256, 0, true, tid, wave * 32, 0);
    __syncthreads();

    // ---- rgb = sigmoid(y @ rw + rb) * 1.002 - 0.001 ----
    if (tid < 96) {
        int row = tid / 3, ch = tid % 3;
        float acc = rb[ch];
        for (int k = 0; k < 256; ++k)
            acc = fmaf(sBufY[row * 256 + k], rw[k * 3 + ch], acc);
        float sg = 1.f / (1.f + expf(-acc));
        out[N_RAYS + (size_t)(rayBase + row) * 3 + ch] = sg * 1.002f - 0.001f;
    }
}

// ======================================================================
extern "C" void kernel_launch(void* const* d_in, const int* in_sizes, int n_in,
                              void* d_out, int out_size, void* d_ws, size_t ws_size,
                              hipStream_t stream) {
    const float* means      = (const float*)d_in[0];
    const float* stds       = (const float*)d_in[1];
    const float* viewdirs   = (const float*)d_in[2];
    const float* embeddings = (const float*)d_in[3];
    const float* dw0 = (const float*)d_in[4];
    const float* db0 = (const float*)d_in[5];
    const float* dw1 = (const float*)d_in[6];
    const float* db1 = (const float*)d_in[7];
    const float* sw0 = (const float*)d_in[8];
    const float* sb0 = (const float*)d_in[9];
    const float* sw1 = (const float*)d_in[10];
    const float* sb1 = (const float*)d_in[11];
    const float* rw  = (const float*)d_in[12];
    const float* rb  = (const float*)d_in[13];
    float* out  = (float*)d_out;
    float* feat = (float*)d_ws;   // [N_RAYS, 40] scratch

    {
        int total = N_RAYS * NUM_LEVELS;
        grid_encode_kernel<<<dim3((total + 255) / 256), dim3(256), 0, stream>>>(
            means, stds, embeddings, feat);
    }
    {
        const int smemBytes = SMEM_FLOATS * (int)sizeof(float);   // 140,304 B
        (void)hipFuncSetAttribute((const void*)nerf_mlp_kernel,
                                  hipFuncAttributeMaxDynamicSharedMemorySize,
                                  smemBytes);
        nerf_mlp_kernel<<<dim3(N_RAYS / 32), dim3(256), smemBytes, stream>>>(
            feat, viewdirs, dw0, db0, dw1, db1, sw0, sb0, sw1, sb1, rw, rb, out);
    }
}